// convGCN_14602888806748
// MI455X (gfx1250) — compile-verified
//
#include <hip/hip_runtime.h>

typedef __attribute__((ext_vector_type(2))) float v2f;
typedef __attribute__((ext_vector_type(8))) float v8f;

#define R_DIM 64      // rank of eg_vectors
#define C_DIM 256     // IN_DIM == OUT_DIM

// ---------------------------------------------------------------------------
// Kernel 0: zero the atomic-accumulated scratch (u[64] + proj[64*256]).
// ---------------------------------------------------------------------------
__global__ void zero_kernel(float* __restrict__ p, int n) {
    int i = blockIdx.x * blockDim.x + threadIdx.x;
    if (i < n) p[i] = 0.0f;
}

// ---------------------------------------------------------------------------
// Kernel 1: u[r] = sum_n eg[n][r].  Block-partial sums + LDS reduce + atomics.
// ---------------------------------------------------------------------------
__global__ void reduce_u_kernel(const float* __restrict__ eg,
                                float* __restrict__ u, int n) {
    __shared__ float smem[256];
    int tid  = threadIdx.x;
    int col  = tid & 63;      // r index
    int rgrp = tid >> 6;      // 0..3 row groups per block
    float s = 0.0f;
    for (int row = blockIdx.x * 4 + rgrp; row < n; row += gridDim.x * 4)
        s += eg[row * R_DIM + col];
    smem[tid] = s;
    __syncthreads();
    if (tid < 64) {
        float v = smem[tid] + smem[tid + 64] + smem[tid + 128] + smem[tid + 192];
        atomicAdd(&u[tid], v);
    }
}

// ---------------------------------------------------------------------------
// Kernel 2: proj[64][256] += eg^T @ x over a contiguous row-chunk per block.
// 256 threads = 8 waves; wave w owns row-tile (w&3) and 8 col-tiles.
// K-pair-interleaved LDS layouts: fragment (k0,k0+1) for a lane is 8
// contiguous, 8B-aligned bytes -> single ds_load_b64, no register shuffling.
// Pair-row strides == 32 (mod 64 banks) so lane halves use disjoint banks.
// ---------------------------------------------------------------------------
#define EGP 160   // floats per eg pair-row (2*64 + 32); 160 % 64 == 32
#define XP  544   // floats per x  pair-row (2*256 + 32); 544 % 64 == 32

__global__ void proj_kernel(const float* __restrict__ eg,
                            const float* __restrict__ x,
                            float* __restrict__ proj, int n) {
    __shared__ float egp[16 * EGP];   // 16 k-pairs x (64 r, parity)
    __shared__ float xp[16 * XP];     // 16 k-pairs x (256 c, parity)

    int tid   = threadIdx.x;
    int lane  = tid & 31;
    int w     = tid >> 5;
    int mrow  = lane & 15;
    int khalf = lane >> 4;        // 0 or 1 (K offset 0 / 2)
    int rt    = w & 3;            // row tile (r dimension)
    int ctb   = (w >> 2) * 8;     // first of 8 col tiles

    int chunk = (n + gridDim.x - 1) / gridDim.x;
    int n0    = blockIdx.x * chunk;
    int nend  = min(n0 + chunk, n);

    v8f acc[8];
#pragma unroll
    for (int t = 0; t < 8; ++t)
#pragma unroll
        for (int j = 0; j < 8; ++j) acc[t][j] = 0.0f;

    for (int nb = n0; nb < nend; nb += 32) {
        // stage 32 rows of eg, pair-interleaved: egp[k>>1][col*2 + (k&1)]
#pragma unroll
        for (int i = 0; i < 8; ++i) {
            int row = i * 4 + (tid >> 6);
            int col = tid & 63;
            int g   = nb + row;
            float v = (g < nend) ? eg[g * R_DIM + col] : 0.0f;
            egp[(row >> 1) * EGP + col * 2 + (row & 1)] = v;
        }
        // stage 32 rows of x, pair-interleaved: xp[k>>1][col*2 + (k&1)]
#pragma unroll
        for (int i = 0; i < 32; ++i) {
            int g   = nb + i;
            float v = (g < nend) ? x[g * C_DIM + tid] : 0.0f;
            xp[(i >> 1) * XP + tid * 2 + (i & 1)] = v;
        }
        __syncthreads();

#pragma unroll
        for (int kk = 0; kk < 8; ++kk) {        // 32 rows of K, 4 per WMMA
            int pidx = kk * 2 + khalf;          // K pair index (k0 = 2*pidx)
            // A = eg^T : A[m=r][k=n]; one aligned b64 LDS read per lane
            v2f a = *(const v2f*)&egp[pidx * EGP + (rt * 16 + mrow) * 2];
#pragma unroll
            for (int t = 0; t < 8; ++t) {
                int c = (ctb + t) * 16 + mrow;
                // B = x : B[k=n][c]; one aligned b64 LDS read per lane
                v2f b = *(const v2f*)&xp[pidx * XP + c * 2];
                acc[t] = __builtin_amdgcn_wmma_f32_16x16x4_f32(
                    false, a, false, b, (short)0, acc[t], false, false);
            }
        }
        __syncthreads();
    }

    // accumulate partial [64x256] into global proj
#pragma unroll
    for (int t = 0; t < 8; ++t) {
        int c = (ctb + t) * 16 + mrow;
#pragma unroll
        for (int i = 0; i < 8; ++i) {
            int m = i + 8 * khalf;              // C layout: M = vgpr + 8*half
            int r = rt * 16 + m;
            atomicAdd(&proj[r * C_DIM + c], acc[t][i]);
        }
    }
}

// ---------------------------------------------------------------------------
// Kernel 3: P2 = proj @ W.  Tiny (8.4 MFLOP); one workgroup.
// Stores P2 pair-interleaved over its row (=K for the output GEMM):
//   P2i[(k>>1)*512 + c*2 + (k&1)]  -> out_kernel B fragment = one b64 load.
// ---------------------------------------------------------------------------
__global__ void p2_kernel(const float* __restrict__ proj,
                          const float* __restrict__ W,
                          float* __restrict__ P2i) {
    int tid   = threadIdx.x;
    int lane  = tid & 31;
    int w     = tid >> 5;
    int mrow  = lane & 15;
    int khalf = lane >> 4;
    int rt    = w & 3;
    int ctb   = (w >> 2) * 8;

    v8f acc[8];
#pragma unroll
    for (int t = 0; t < 8; ++t)
#pragma unroll
        for (int j = 0; j < 8; ++j) acc[t][j] = 0.0f;

    for (int kk = 0; kk < 64; ++kk) {           // K = 256
        int k0 = kk * 4 + 2 * khalf;            // even
        // A = proj[m=r][k0..k0+1]: contiguous -> global b64 load
        v2f a = *(const v2f*)&proj[(rt * 16 + mrow) * C_DIM + k0];
#pragma unroll
        for (int t = 0; t < 8; ++t) {
            int c0 = (ctb + t) * 16;
            v2f b;                              // B = W[k][c] (row-major input)
            b.x = W[k0 * C_DIM + c0 + mrow];
            b.y = W[(k0 + 1) * C_DIM + c0 + mrow];
            acc[t] = __builtin_amdgcn_wmma_f32_16x16x4_f32(
                false, a, false, b, (short)0, acc[t], false, false);
        }
    }
#pragma unroll
    for (int t = 0; t < 8; ++t) {
        int c = (ctb + t) * 16 + mrow;
#pragma unroll
        for (int i = 0; i < 8; ++i) {
            int r = rt * 16 + i + 8 * khalf;    // P2 row (K of output GEMM)
            P2i[(r >> 1) * (2 * C_DIM) + c * 2 + (r & 1)] = acc[t][i];
        }
    }
}

// ---------------------------------------------------------------------------
// Kernel 4: out[n][c] = (1/deg[n]) * (eg[n,:] @ P2)[c] + bias[c]
// 64 rows per workgroup; deg computed on the fly from u; fused scale+bias.
// A fragment: aligned ds_load_b64 (k0,k0+1 contiguous in egLDS row).
// B fragment: aligned global_load_b64 from pair-interleaved P2 (L0/L2-hot).
// ---------------------------------------------------------------------------
#define EGO 68    // LDS row stride: b64 A reads conflict-free (68 % 64 == 4)

__global__ void out_kernel(const float* __restrict__ eg,
                           const float* __restrict__ P2i,
                           const float* __restrict__ u,
                           const float* __restrict__ bias,
                           float* __restrict__ out, int n) {
    __shared__ float egLDS[64 * EGO];
    __shared__ float uLDS[64];
    __shared__ float invLDS[64];
    __shared__ float biasLDS[256];

    int tid = threadIdx.x;
    int rb  = blockIdx.x * 64;

    if (tid < 64) uLDS[tid] = u[tid];
    biasLDS[tid] = bias[tid];
#pragma unroll
    for (int i = 0; i < 16; ++i) {
        int row = i * 4 + (tid >> 6);
        int col = tid & 63;
        int g   = rb + row;
        egLDS[row * EGO + col] = (g < n) ? eg[g * R_DIM + col] : 0.0f;
    }
    __syncthreads();

    if (tid < 64) {                             // deg = eg_row . u ; inv = 1/deg
        float s = 0.0f;
#pragma unroll
        for (int r = 0; r < R_DIM; ++r) s += egLDS[tid * EGO + r] * uLDS[r];
        invLDS[tid] = 1.0f / s;
    }
    __syncthreads();

    int lane  = tid & 31;
    int w     = tid >> 5;
    int mrow  = lane & 15;
    int khalf = lane >> 4;
    int rt    = w & 3;
    int ctb   = (w >> 2) * 8;

    const v2f* P2v = (const v2f*)P2i;           // P2v[(k>>1)*256 + c] = {P2[k][c], P2[k+1][c]}

    v8f acc[8];
#pragma unroll
    for (int t = 0; t < 8; ++t)
#pragma unroll
        for (int j = 0; j < 8; ++j) acc[t][j] = 0.0f;

#pragma unroll
    for (int kk = 0; kk < 16; ++kk) {           // K = 64
        int k0 = kk * 4 + 2 * khalf;            // even
        // A = eg[m=row][k0..k0+1]: contiguous, 8B-aligned -> ds_load_b64
        v2f a = *(const v2f*)&egLDS[(rt * 16 + mrow) * EGO + k0];
#pragma unroll
        for (int t = 0; t < 8; ++t) {
            int c = (ctb + t) * 16 + mrow;
            v2f b = P2v[(k0 >> 1) * C_DIM + c];     // one global b64 load
            acc[t] = __builtin_amdgcn_wmma_f32_16x16x4_f32(
                false, a, false, b, (short)0, acc[t], false, false);
        }
    }

#pragma unroll
    for (int t = 0; t < 8; ++t) {
        int c = (ctb + t) * 16 + mrow;
#pragma unroll
        for (int i = 0; i < 8; ++i) {
            int m    = i + 8 * khalf;
            int grow = rb + rt * 16 + m;
            if (grow < n)
                out[grow * C_DIM + c] =
                    acc[t][i] * invLDS[rt * 16 + m] + biasLDS[c];
        }
    }
}

// ---------------------------------------------------------------------------
// Launch: inputs = {x, edge_index(unused), eg_vectors, weights, bias}
// ---------------------------------------------------------------------------
extern "C" void kernel_launch(void* const* d_in, const int* in_sizes, int n_in,
                              void* d_out, int out_size, void* d_ws, size_t ws_size,
                              hipStream_t stream) {
    const float* x    = (const float*)d_in[0];
    const float* eg   = (const float*)d_in[2];
    const float* W    = (const float*)d_in[3];
    const float* bias = (const float*)d_in[4];
    float*       out  = (float*)d_out;

    int N = in_sizes[2] / R_DIM;                // eg_vectors is [N, 64]

    float* ws   = (float*)d_ws;
    float* u    = ws;                           // [64]
    float* proj = ws + 64;                      // [64*256]
    float* P2i  = ws + 64 + R_DIM * C_DIM;      // [64*256] pair-interleaved

    int nz = 64 + R_DIM * C_DIM;                // only atomic targets need zeroing
    zero_kernel<<<(nz + 255) / 256, 256, 0, stream>>>(ws, nz);
    reduce_u_kernel<<<512, 256, 0, stream>>>(eg, u, N);
    proj_kernel<<<250, 256, 0, stream>>>(eg, x, proj, N);
    p2_kernel<<<1, 256, 0, stream>>>(proj, W, P2i);
    out_kernel<<<(N + 63) / 64, 256, 0, stream>>>(eg, P2i, u, bias, out, N);
}